// ensemble_linear_5059471475117
// MI455X (gfx1250) — compile-verified
//
#include <hip/hip_runtime.h>
#include <hip/hip_bf16.h>

// Block-diagonal ensemble linear: y[b, n*100+o] = sum_i W[n*100+o, n*200+i] * x[b, n*200+i]
// 20 independent (32768x200)x(200x100) fp32 GEMMs via V_WMMA_F32_16X16X4_F32.
// W block staged in LDS pre-swizzled to WMMA B layout in [k][tile][lane] order
// so all 7 N-tile operands of a K-step sit within ds_load_2addr_b64 offset reach.
// One WMMA per accumulator per loop iteration (7 independent D->C chains).
// x loads / y stores are non-temporal (single-pass streams; keep L2 for W).

typedef __attribute__((ext_vector_type(2))) float v2f;
typedef __attribute__((ext_vector_type(8))) float v8f;

#define BATCH      32768
#define INPUT_SZ   200
#define OUTPUT_SZ  100
#define NUM_BLK    20
#define XCOLS      (INPUT_SZ * NUM_BLK)   // 4000
#define YCOLS      (OUTPUT_SZ * NUM_BLK)  // 2000
#define NTILES     7                      // ceil(100/16) N tiles
#define NSTEP      50                     // 200 / 4 K-steps
#define LDS_ENTRIES (NSTEP * NTILES * 32) // 11200 x float2 = 89600 B
#define GROUPS_PER_N 64                   // 2048 M-tiles / (8 waves * 4 per wave)
#define MT_PER_WAVE  4

__global__ __launch_bounds__(256) void ensemble_linear_wmma_f32(
    const float* __restrict__ x,
    const float* __restrict__ W,
    float* __restrict__ y)
{
    extern __shared__ v2f ldsB[];          // [kstep][tile][lane] pre-swizzled B

    const int tid    = threadIdx.x;
    const int lane   = tid & 31;
    const int wave   = tid >> 5;
    const int laneLo = lane & 15;
    const int half   = lane >> 4;          // 0: lanes 0-15, 1: lanes 16-31

    const int n     = blockIdx.x / GROUPS_PER_N;   // diagonal block 0..19
    const int group = blockIdx.x % GROUPS_PER_N;   // M-tile group within block n

    // ================= Stage W block n into LDS (pre-swizzled) ===========
    // Entry (k, t, l): v2f = B VGPR0/1 for K-step k, N-tile t, i.e.
    //   { W[o, 4k + 2*half], W[o, 4k + 1 + 2*half] },
    // with o = 16t + (l&15) (clamped), half = l>>4.
    const float* Wblk = W + (size_t)n * OUTPUT_SZ * XCOLS + n * INPUT_SZ;
    for (int idx = tid; idx < LDS_ENTRIES; idx += 256) {
        const int l  = idx & 31;
        const int kt = idx >> 5;
        const int t  = kt % NTILES;
        const int k  = kt / NTILES;
        int o = (t << 4) + (l & 15);
        if (o > OUTPUT_SZ - 1) o = OUTPUT_SZ - 1;  // clamp (stores guarded later)
        const int h = l >> 4;
        ldsB[idx] = *(const v2f*)(Wblk + (size_t)o * XCOLS + (k << 2) + (h << 1));
    }
    __syncthreads();

    // ================= Main loop: 4 M-tiles per wave ======================
    const float* xbase = x + n * INPUT_SZ + 2 * half;
    float* ybase = y + n * OUTPUT_SZ;
    const v2f* ldsLane = ldsB + lane;

    for (int g = 0; g < MT_PER_WAVE; ++g) {
        const int mTile = (group << 5) + (g << 3) + wave;
        const int mBase = mTile << 4;
        const float* xrow = xbase + (size_t)(mBase + laneLo) * XCOLS;

        v8f acc[NTILES];
#pragma unroll
        for (int t = 0; t < NTILES; ++t) acc[t] = (v8f)(0.0f);

        for (int k = 0; k < NSTEP; ++k) {
            // A for K-step k (single-pass stream -> non-temporal)
            v2f a = __builtin_nontemporal_load((const v2f*)(xrow + (k << 2)));
            const v2f* bk = ldsLane + (size_t)k * NTILES * 32;
#pragma unroll
            for (int t = 0; t < NTILES; ++t) {
                v2f b = bk[t * 32];                        // 7 x 256B apart: 2addr reach
                acc[t] = __builtin_amdgcn_wmma_f32_16x16x4_f32(
                    false, a, false, b, (short)0, acc[t], false, false);
            }
        }

        // Store D: VGPR r -> M = mBase + r + 8*half, N = 16*t + laneLo
#pragma unroll
        for (int t = 0; t < NTILES; ++t) {
            const int o = (t << 4) + laneLo;
            if (o < OUTPUT_SZ) {
#pragma unroll
                for (int r = 0; r < 8; ++r) {
                    const int M = mBase + r + 8 * half;
                    __builtin_nontemporal_store(acc[t][r],
                        ybase + (size_t)M * YCOLS + o);
                }
            }
        }
    }
}

extern "C" void kernel_launch(void* const* d_in, const int* in_sizes, int n_in,
                              void* d_out, int out_size, void* d_ws, size_t ws_size,
                              hipStream_t stream) {
    const float* x = (const float*)d_in[0];  // (32768, 4000) fp32
    const float* W = (const float*)d_in[1];  // (2000, 4000) fp32
    float* y = (float*)d_out;                // (32768, 2000) fp32

    const int numBlocks = NUM_BLK * GROUPS_PER_N;              // 1280
    const size_t ldsBytes = (size_t)LDS_ENTRIES * sizeof(v2f); // 89600 B < 320 KB
    ensemble_linear_wmma_f32<<<numBlocks, 256, ldsBytes, stream>>>(x, W, y);
}